// MultiBoxLoss_10823317586364
// MI455X (gfx1250) — compile-verified
//
#include <hip/hip_runtime.h>
#include <hip/hip_bf16.h>
#include <math.h>

// Problem constants (match reference)
#define NN 64
#define PP 8732
#define CC 81
#define MM 64
#define SORT_SZ 16384   // next pow2 >= PP
#define THRESH 0.5f
#define NPR 3           // neg:pos ratio

typedef __attribute__((ext_vector_type(2))) float v2f;
typedef __attribute__((ext_vector_type(8))) float v8f;

// ---------------------------------------------------------------------------
// Kernel 1: per-image IoU matching, target encoding, positive count + L1 sum.
// One block per image; GT boxes cached in LDS. Deterministic tree reduction.
// ---------------------------------------------------------------------------
__global__ __launch_bounds__(256) void mbl_match(
    const float* __restrict__ plocs, const float* __restrict__ boxes,
    const int* __restrict__ labels, const int* __restrict__ nobj,
    const float* __restrict__ priors,
    int* __restrict__ tc_out, float* __restrict__ l1sum_out, int* __restrict__ npos_out)
{
    __shared__ float bx[MM][4];     // xyxy
    __shared__ float bc[MM][4];     // cxcywh
    __shared__ float barea[MM];
    __shared__ int   blab[MM];
    __shared__ float redf[256];
    __shared__ int   redi[256];

    const int n   = blockIdx.x;
    const int tid = threadIdx.x;
    const int nval = nobj[n];

    for (int m = tid; m < MM; m += blockDim.x) {
        const float* b = boxes + ((size_t)n * MM + m) * 4;
        float x1 = b[0], y1 = b[1], x2 = b[2], y2 = b[3];
        bx[m][0] = x1; bx[m][1] = y1; bx[m][2] = x2; bx[m][3] = y2;
        bc[m][0] = (x1 + x2) * 0.5f; bc[m][1] = (y1 + y2) * 0.5f;
        bc[m][2] = x2 - x1;          bc[m][3] = y2 - y1;
        barea[m] = (x2 - x1) * (y2 - y1);
        blab[m]  = labels[(size_t)n * MM + m];
    }
    __syncthreads();

    float l1_local = 0.f;
    int   np_local = 0;
    for (int p = tid; p < PP; p += blockDim.x) {
        float pcx = priors[p * 4 + 0], pcy = priors[p * 4 + 1];
        float pw  = priors[p * 4 + 2], ph  = priors[p * 4 + 3];
        float px1 = pcx - pw * 0.5f, py1 = pcy - ph * 0.5f;
        float px2 = pcx + pw * 0.5f, py2 = pcy + ph * 0.5f;
        float pa  = (px2 - px1) * (py2 - py1);

        float best = -2.f; int bi = 0;
        for (int m = 0; m < MM; ++m) {
            float ov;
            if (m < nval) {
                float lx = fmaxf(bx[m][0], px1), ly = fmaxf(bx[m][1], py1);
                float hx = fminf(bx[m][2], px2), hy = fminf(bx[m][3], py2);
                float iw = fmaxf(hx - lx, 0.f),  ih = fmaxf(hy - ly, 0.f);
                float inter = iw * ih;
                ov = inter / (barea[m] + pa - inter);
            } else {
                ov = -1.f;   // matches reference masking of padded objects
            }
            if (ov > best) { best = ov; bi = m; }   // strict > keeps first max (jnp.argmax)
        }
        int tcv = (best < THRESH) ? 0 : blab[bi];
        tc_out[(size_t)n * PP + p] = tcv;
        if (tcv != 0) {
            np_local++;
            float g0 = (bc[bi][0] - pcx) / (pw * 0.1f);
            float g1 = (bc[bi][1] - pcy) / (ph * 0.1f);
            float g2 = logf(bc[bi][2] / pw) * 5.f;
            float g3 = logf(bc[bi][3] / ph) * 5.f;
            const float* pl = plocs + ((size_t)n * PP + p) * 4;
            l1_local += fabsf(pl[0] - g0) + fabsf(pl[1] - g1)
                      + fabsf(pl[2] - g2) + fabsf(pl[3] - g3);
        }
    }
    redf[tid] = l1_local; redi[tid] = np_local;
    __syncthreads();
    for (int s = 128; s; s >>= 1) {
        if (tid < s) { redf[tid] += redf[tid + s]; redi[tid] += redi[tid + s]; }
        __syncthreads();
    }
    if (tid == 0) { l1sum_out[n] = redf[0]; npos_out[n] = redi[0]; }
}

// ---------------------------------------------------------------------------
// Kernel 2: streaming cross-entropy. One wave (32 lanes) per prior: coalesced
// read of the 81 contiguous class scores, cross-lane max + exp-sum (wave32
// shuffles), conf = lse - score[true_class]. global_prefetch hides latency.
// This is the kernel that touches the 181 MB tensor -> HBM-bound (~8 us floor).
// ---------------------------------------------------------------------------
__global__ __launch_bounds__(256) void mbl_conf(
    const float* __restrict__ scores, const int* __restrict__ tc,
    float* __restrict__ conf_all)
{
    const int lane      = threadIdx.x & 31;
    const int waveInBlk = threadIdx.x >> 5;
    const long long wid   = (long long)blockIdx.x * (blockDim.x >> 5) + waveInBlk;
    const long long nw    = (long long)gridDim.x * (blockDim.x >> 5);
    const long long total = (long long)NN * PP;

    for (long long idx = wid; idx < total; idx += nw) {
        const float* base = scores + idx * CC;
        long long nxt = idx + nw;
        if (nxt < total) {
            // gfx1250: lowers to global_prefetch_b8; lanes cover the 324B row
            __builtin_prefetch(scores + nxt * CC + (long long)lane * 4, 0, 0);
        }
        float v0 = (lane      < CC) ? base[lane]      : -__builtin_inff();
        float v1 = (lane + 32 < CC) ? base[lane + 32] : -__builtin_inff();
        float v2 = (lane + 64 < CC) ? base[lane + 64] : -__builtin_inff();
        float m = fmaxf(fmaxf(v0, v1), v2);
        for (int off = 16; off; off >>= 1) m = fmaxf(m, __shfl_xor(m, off));
        float s = 0.f;
        if (lane      < CC) s += expf(v0 - m);
        if (lane + 32 < CC) s += expf(v1 - m);
        if (lane + 64 < CC) s += expf(v2 - m);
        for (int off = 16; off; off >>= 1) s += __shfl_xor(s, off);
        float lse = m + logf(s);
        int t = tc[idx];
        float stc = base[t];           // broadcast load (cache-resident)
        if (lane == 0) conf_all[idx] = lse - stc;
    }
}

// ---------------------------------------------------------------------------
// Kernel 3: per-image hard-negative mining. One 1024-thread block per image.
// (a) build conf_neg in LDS + deterministic positive-CE sum,
// (b) bitonic sort 16384 f32 descending in LDS,
// (c) top-K masked sum via V_WMMA_F32_16X16X4_F32: A = 16x4 tile of masked
//     sorted values, B = ones -> accumulator columns hold row sums; column 0
//     (lanes 0 and 16) yields the wave total. Full f32 precision, so this is
//     numerically a plain summation, executed on the matrix pipe.
// LDS reads are hoisted out of the mask select (always in-bounds) so the
// compiler emits unconditional ds_load_b64 + v_cndmask instead of
// exec-predicated loads around each WMMA.
// ---------------------------------------------------------------------------
__global__ __launch_bounds__(1024) void mbl_sort_topk(
    const float* __restrict__ conf_all, const int* __restrict__ tc,
    const int* __restrict__ npos,
    float* __restrict__ possum_out, float* __restrict__ hardsum_out)
{
    __shared__ float sd[SORT_SZ];   // 64 KB (CDNA5 WGP has 320 KB LDS)
    const int n   = blockIdx.x;
    const int tid = threadIdx.x;
    const int nt  = blockDim.x;     // 1024

    // (a) load conf_neg, accumulate positive CE
    float pos_local = 0.f;
    for (int p = tid; p < PP; p += nt) {
        float c = conf_all[(size_t)n * PP + p];
        int   t = tc[(size_t)n * PP + p];
        sd[p] = (t != 0) ? 0.f : c;
        if (t != 0) pos_local += c;
    }
    sd[PP + tid] = pos_local;               // stash partials in pad region
    __syncthreads();
    for (int s = 512; s; s >>= 1) {
        if (tid < s) sd[PP + tid] += sd[PP + tid + s];
        __syncthreads();
    }
    if (tid == 0) possum_out[n] = sd[PP];
    __syncthreads();
    for (int i = PP + tid; i < SORT_SZ; i += nt) sd[i] = -1.f;  // pad below all CE (>=0)
    __syncthreads();

    // (b) bitonic sort, descending
    for (int k = 2; k <= SORT_SZ; k <<= 1) {
        for (int j = k >> 1; j > 0; j >>= 1) {
            for (int i = tid; i < SORT_SZ; i += nt) {
                int l = i ^ j;
                if (l > i) {
                    float a = sd[i], b = sd[l];
                    bool up = ((i & k) == 0);
                    if (up == (a < b)) { sd[i] = b; sd[l] = a; }
                }
            }
            __syncthreads();
        }
    }

    // (c) top-K sum on the matrix pipe
    int K = npos[n] * NPR;
    if (K > PP) K = PP;

    const int lane = tid & 31;
    const int wave = tid >> 5;              // 0..31, each owns 512 elements
    v8f acc = {};
    v2f bones = {1.f, 1.f};
    const int base = wave * (SORT_SZ / 32);
    #pragma unroll
    for (int it = 0; it < (SORT_SZ / 32) / 64; ++it) {   // 8 x 64-element tiles
        const int i0 = base + it * 64 + lane * 2;
        // Unconditional LDS reads (i0+1 < SORT_SZ always) -> ds_load_b64;
        // top-K mask becomes a v_cndmask select, EXEC stays all-ones.
        float x0 = sd[i0];
        float x1 = sd[i0 + 1];
        v2f a;
        a[0] = (i0     < K) ? x0 : 0.f;     // mask: arange < K
        a[1] = (i0 + 1 < K) ? x1 : 0.f;
        acc = __builtin_amdgcn_wmma_f32_16x16x4_f32(
            /*neg_a=*/false, a, /*neg_b=*/false, bones,
            /*c_mod=*/(short)0, acc, /*reuse_a=*/false, /*reuse_b=*/false);
    }
    float rowTot  = acc[0] + acc[1] + acc[2] + acc[3] + acc[4] + acc[5] + acc[6] + acc[7];
    float waveTot = __shfl(rowTot, 0) + __shfl(rowTot, 16);   // column 0 = M rows 0..7 + 8..15
    __syncthreads();                        // all waves done reading sd
    if (lane == 0) sd[wave] = waveTot;
    __syncthreads();
    if (tid == 0) {
        float h = 0.f;
        for (int w = 0; w < 32; ++w) h += sd[w];
        hardsum_out[n] = h;
    }
}

// ---------------------------------------------------------------------------
// Kernel 4: combine per-image partials into the scalar loss.
// ---------------------------------------------------------------------------
__global__ __launch_bounds__(64) void mbl_final(
    const float* __restrict__ l1sum, const float* __restrict__ possum,
    const float* __restrict__ hardsum, const int* __restrict__ npos,
    float* __restrict__ out)
{
    __shared__ float rf[64], rp[64], rh[64];
    __shared__ int   ri[64];
    int t = threadIdx.x;
    rf[t] = l1sum[t]; rp[t] = possum[t]; rh[t] = hardsum[t]; ri[t] = npos[t];
    __syncthreads();
    for (int s = 32; s; s >>= 1) {
        if (t < s) { rf[t] += rf[t+s]; rp[t] += rp[t+s]; rh[t] += rh[t+s]; ri[t] += ri[t+s]; }
        __syncthreads();
    }
    if (t == 0) {
        float npt  = (float)ri[0];
        float loc  = rf[0] / (npt * 4.f);
        float conf = (rh[0] + rp[0]) / npt;
        out[0] = conf + 1.0f * loc;   // ALPHA = 1
    }
}

extern "C" void kernel_launch(void* const* d_in, const int* in_sizes, int n_in,
                              void* d_out, int out_size, void* d_ws, size_t ws_size,
                              hipStream_t stream) {
    (void)in_sizes; (void)n_in; (void)out_size; (void)ws_size;
    const float* plocs  = (const float*)d_in[0];   // (N,P,4) f32
    const float* scores = (const float*)d_in[1];   // (N,P,C) f32
    const float* boxes  = (const float*)d_in[2];   // (N,M,4) f32
    const int*   labels = (const int*)d_in[3];     // (N,M)   i32
    const int*   nobj   = (const int*)d_in[4];     // (N,)    i32
    const float* priors = (const float*)d_in[5];   // (P,4)   f32
    float* out = (float*)d_out;

    // workspace carve (~4.5 MB)
    int*   tc       = (int*)d_ws;                          // N*P
    float* conf_all = (float*)(tc + (size_t)NN * PP);      // N*P
    float* l1sum    = conf_all + (size_t)NN * PP;          // N
    int*   npos     = (int*)(l1sum + NN);                  // N
    float* possum   = (float*)(npos + NN);                 // N
    float* hardsum  = possum + NN;                         // N

    mbl_match    <<<NN,   256, 0, stream>>>(plocs, boxes, labels, nobj, priors, tc, l1sum, npos);
    mbl_conf     <<<4096, 256, 0, stream>>>(scores, tc, conf_all);
    mbl_sort_topk<<<NN,  1024, 0, stream>>>(conf_all, tc, npos, possum, hardsum);
    mbl_final    <<<1,     64, 0, stream>>>(l1sum, possum, hardsum, npos, out);
}